// GNNEncoder_83056077570929
// MI455X (gfx1250) — compile-verified
//
#include <hip/hip_runtime.h>

typedef float v2f __attribute__((ext_vector_type(2)));
typedef float v8f __attribute__((ext_vector_type(8)));

#define LEAKY_SLOPE 0.2f

// ---------------------------------------------------------------------------
// Monotone float<->uint encoding so segment-max uses native u32 atomic max.
// ---------------------------------------------------------------------------
__device__ __forceinline__ unsigned enc_f32(float v) {
  unsigned u = __float_as_uint(v);
  return (u >> 31) ? ~u : (u | 0x80000000u);
}
__device__ __forceinline__ float dec_f32(unsigned u) {
  return (u >> 31) ? __uint_as_float(u ^ 0x80000000u) : __uint_as_float(~u);
}

// ---------------------------------------------------------------------------
// fp32 WMMA GEMM:  C[M,N] = A[M,K] @ B[K,N] + bias, optional ReLU.
// REQUIRES: M % (16*MT) == 0, N % (16*NT) == 0.  K arbitrary (tail once).
// One wave computes a (16*MT) x (16*NT) output block; blockDim = 256.
// V_WMMA_F32_16X16X4_F32 lane layouts (ISA 7.12.2):
//   A 16x4 : lanes 0-15 rows 0-15, K={k0,k0+1}; lanes 16-31 K={k0+2,k0+3}
//   B 4x16 : lanes 0-15 cols 0-15, K={k0,k0+1}; lanes 16-31 K={k0+2,k0+3}
//   C 16x16: VGPR r -> row (half*8 + r), col = lane%16
// Per k-step: all A/B fragments loaded first (one clause + one wait), then
// MT*NT independent back-to-back WMMAs (hides accumulate latency; B reused
// across MT row-tiles, A reused across NT col-tiles -> 1.5 loads/WMMA).
// ---------------------------------------------------------------------------
template <int MT, int NT>
__global__ void wmma_gemm_f32(const float* __restrict__ A,
                              const float* __restrict__ B,
                              const float* __restrict__ bias,
                              float* __restrict__ C,
                              int M, int N, int K, int relu)
{
  const int tiles_n = N / (16 * NT);
  const int total   = (M / (16 * MT)) * tiles_n;
  const int tile    = blockIdx.x * (blockDim.x >> 5) + (threadIdx.x >> 5);
  if (tile >= total) return;                 // wave-uniform: EXEC stays full
  const int tm   = (tile / tiles_n) * (16 * MT);
  const int tn   = (tile % tiles_n) * (16 * NT);
  const int lane = threadIdx.x & 31;
  const int half = lane >> 4;
  const int l16  = lane & 15;

  // per-lane base pointers; loop index math stays scalar
  const float* Ap = A + (long)(tm + l16) * K + (half << 1);
  const float* Bp = B + (long)(half << 1) * N + tn + l16;
  const long   arow = (long)16 * K;          // A stride between row-tiles

  v8f acc[MT][NT];
#pragma unroll
  for (int m = 0; m < MT; ++m)
#pragma unroll
    for (int t = 0; t < NT; ++t)
      acc[m][t] = (v8f){0.f, 0.f, 0.f, 0.f, 0.f, 0.f, 0.f, 0.f};

  const int K4 = K & ~3;
  for (int k0 = 0; k0 < K4; k0 += 4) {
    v2f a[MT];
#pragma unroll
    for (int m = 0; m < MT; ++m) {
      a[m].x = Ap[m * arow + k0];
      a[m].y = Ap[m * arow + k0 + 1];
    }
    const float* bk = Bp + (long)k0 * N;
    v2f b[NT];
#pragma unroll
    for (int t = 0; t < NT; ++t) {
      b[t].x = bk[16 * t];
      b[t].y = bk[16 * t + N];
    }
#pragma unroll
    for (int m = 0; m < MT; ++m)
#pragma unroll
      for (int t = 0; t < NT; ++t)
        acc[m][t] = __builtin_amdgcn_wmma_f32_16x16x4_f32(
            false, a[m], false, b[t], (short)0, acc[m][t], false, false);
  }
  if (K4 < K) {                              // K remainder (1..3), once
    const int ak = K4 + (half << 1);
    v2f a[MT];
#pragma unroll
    for (int m = 0; m < MT; ++m) {
      a[m].x = (ak < K)     ? Ap[m * arow + K4]     : 0.f;
      a[m].y = (ak + 1 < K) ? Ap[m * arow + K4 + 1] : 0.f;
    }
    const float* bk = Bp + (long)K4 * N;
    v2f b[NT];
#pragma unroll
    for (int t = 0; t < NT; ++t) {
      b[t].x = (ak < K)     ? bk[16 * t]     : 0.f;
      b[t].y = (ak + 1 < K) ? bk[16 * t + N] : 0.f;
    }
#pragma unroll
    for (int m = 0; m < MT; ++m)
#pragma unroll
      for (int t = 0; t < NT; ++t)
        acc[m][t] = __builtin_amdgcn_wmma_f32_16x16x4_f32(
            false, a[m], false, b[t], (short)0, acc[m][t], false, false);
  }

  // store: lane half owns rows (tm + m*16 + 8*half + r), col tn + 16*t + l16
#pragma unroll
  for (int m = 0; m < MT; ++m) {
    float* Cp = C + (long)(tm + m * 16 + (half << 3)) * N + tn + l16;
#pragma unroll
    for (int t = 0; t < NT; ++t) {
      const float bv = bias ? bias[tn + 16 * t + l16] : 0.f;
#pragma unroll
      for (int r = 0; r < 8; ++r) {
        float v = acc[m][t][r] + bv;
        if (relu) v = fmaxf(v, 0.f);
        Cp[(long)r * N + 16 * t] = v;
      }
    }
  }
}

// ---------------------------------------------------------------------------
// Zero-fill (also used for encoded max buffers: 0u encodes below -inf).
// ---------------------------------------------------------------------------
__global__ void zero_f32(float* __restrict__ p, long n)
{
  long i = (long)blockIdx.x * blockDim.x + threadIdx.x;
  const long stride = (long)gridDim.x * blockDim.x;
  for (; i < n; i += stride) p[i] = 0.f;
}

// ---------------------------------------------------------------------------
// Per-edge attention logits + segment max. One wave per edge; lane owns 4
// consecutive channels (head h = lane>>3 since heads are contiguous blocks
// of 32 channels). eattr/we may be null (conv2 has no edge features).
// ---------------------------------------------------------------------------
__global__ void edge_logits_kernel(const float* __restrict__ xl,    // [Ns,128]
                                   const float* __restrict__ xr,    // [Nd,128]
                                   const float* __restrict__ eattr, // [E,3] | null
                                   const float* __restrict__ we,    // [3,128] | null
                                   const float* __restrict__ att,   // [128]
                                   const int* __restrict__ src,
                                   const int* __restrict__ dst,
                                   float* __restrict__ logits,      // [E,4]
                                   unsigned* __restrict__ maxbuf,   // [Nd,4] encoded
                                   int E)
{
  const int e = blockIdx.x * (blockDim.x >> 5) + (threadIdx.x >> 5);
  if (e >= E) return;
  const int lane = threadIdx.x & 31;
  const int s = src[e], d = dst[e];

  const float4 a = ((const float4*)xl)[(long)s * 32 + lane];
  const float4 b = ((const float4*)xr)[(long)d * 32 + lane];
  float4 m = make_float4(a.x + b.x, a.y + b.y, a.z + b.z, a.w + b.w);

  if (eattr) {
    const float e0 = eattr[(long)e * 3 + 0];
    const float e1 = eattr[(long)e * 3 + 1];
    const float e2 = eattr[(long)e * 3 + 2];
    const float4 w0 = ((const float4*)we)[0 * 32 + lane];
    const float4 w1 = ((const float4*)we)[1 * 32 + lane];
    const float4 w2 = ((const float4*)we)[2 * 32 + lane];
    m.x += e0 * w0.x + e1 * w1.x + e2 * w2.x;
    m.y += e0 * w0.y + e1 * w1.y + e2 * w2.y;
    m.z += e0 * w0.z + e1 * w1.z + e2 * w2.z;
    m.w += e0 * w0.w + e1 * w1.w + e2 * w2.w;
  }

  float4 g;
  g.x = m.x > 0.f ? m.x : LEAKY_SLOPE * m.x;
  g.y = m.y > 0.f ? m.y : LEAKY_SLOPE * m.y;
  g.z = m.z > 0.f ? m.z : LEAKY_SLOPE * m.z;
  g.w = m.w > 0.f ? m.w : LEAKY_SLOPE * m.w;

  const float4 av = ((const float4*)att)[lane];   // att[h*32+c] == att_flat[4*lane]
  float p = g.x * av.x + g.y * av.y + g.z * av.z + g.w * av.w;
  // reduce over the 8 lanes of each head
  p += __shfl_xor(p, 1, 32);
  p += __shfl_xor(p, 2, 32);
  p += __shfl_xor(p, 4, 32);

  if ((lane & 7) == 0) {
    const int h = lane >> 3;
    logits[(long)e * 4 + h] = p;
    atomicMax(&maxbuf[(long)d * 4 + h], enc_f32(p));
  }
}

// ---------------------------------------------------------------------------
// ex = exp(logit - max[dst]); accumulate denominator. Thread per (edge, head).
// In-place overwrite of logits with ex.
// ---------------------------------------------------------------------------
__global__ void edge_exp_kernel(float* __restrict__ logits,          // [E,4]
                                const unsigned* __restrict__ maxbuf, // [Nd,4]
                                float* __restrict__ den,             // [Nd,4]
                                const int* __restrict__ dst, int E)
{
  const int i = blockIdx.x * blockDim.x + threadIdx.x;
  if (i >= E * 4) return;
  const int e = i >> 2, h = i & 3;
  const int d = dst[e];
  const float ex = expf(logits[i] - dec_f32(maxbuf[(long)d * 4 + h]));
  logits[i] = ex;
  atomicAdd(&den[(long)d * 4 + h], ex);
}

// ---------------------------------------------------------------------------
// out[dst] += (ex/den) * xl[src]. One wave per edge, 4 fp32 atomics per lane
// (lands in L2 on MI455X; out buffer fits in 192MB L2).
// ---------------------------------------------------------------------------
__global__ void edge_agg_kernel(const float* __restrict__ xl,  // [Ns,128]
                                const float* __restrict__ exv, // [E,4]
                                const float* __restrict__ den, // [Nd,4]
                                const int* __restrict__ src,
                                const int* __restrict__ dst,
                                float* __restrict__ out,       // [Nd,128]
                                int E)
{
  const int e = blockIdx.x * (blockDim.x >> 5) + (threadIdx.x >> 5);
  if (e >= E) return;
  const int lane = threadIdx.x & 31;
  const int h = lane >> 3;
  const int s = src[e], d = dst[e];
  const float alpha = exv[(long)e * 4 + h] / (den[(long)d * 4 + h] + 1e-16f);
  const float4 v = ((const float4*)xl)[(long)s * 32 + lane];
  float* o = out + (long)d * 128 + lane * 4;
  atomicAdd(o + 0, alpha * v.x);
  atomicAdd(o + 1, alpha * v.y);
  atomicAdd(o + 2, alpha * v.z);
  atomicAdd(o + 3, alpha * v.w);
}

// ---------------------------------------------------------------------------
// In-place bias (+optional ReLU) epilogue over [M,128].
// ---------------------------------------------------------------------------
__global__ void bias_act_kernel(float* __restrict__ p,
                                const float* __restrict__ bias,
                                long n, int relu)
{
  const long i = (long)blockIdx.x * blockDim.x + threadIdx.x;
  if (i >= n) return;
  float v = p[i] + bias[i & 127];
  if (relu) v = fmaxf(v, 0.f);
  p[i] = v;
}

// ===========================================================================
extern "C" void kernel_launch(void* const* d_in, const int* in_sizes, int n_in,
                              void* d_out, int out_size, void* d_ws, size_t ws_size,
                              hipStream_t stream)
{
  const float* customer_x = (const float*)d_in[0];
  const float* fund_x     = (const float*)d_in[1];
  const float* edge_attr  = (const float*)d_in[2];
  const float* user_lin_w = (const float*)d_in[3];
  const float* user_lin_b = (const float*)d_in[4];
  const float* item_lin_w = (const float*)d_in[5];
  const float* item_lin_b = (const float*)d_in[6];
  const float* conv1_wl   = (const float*)d_in[7];
  const float* conv1_bl   = (const float*)d_in[8];
  const float* conv1_wr   = (const float*)d_in[9];
  const float* conv1_br   = (const float*)d_in[10];
  const float* conv1_we   = (const float*)d_in[11];
  const float* conv1_att  = (const float*)d_in[12];
  const float* conv1_bias = (const float*)d_in[13];
  const float* conv2_wl   = (const float*)d_in[14];
  const float* conv2_bl   = (const float*)d_in[15];
  const float* conv2_wr   = (const float*)d_in[16];
  const float* conv2_br   = (const float*)d_in[17];
  const float* conv2_att  = (const float*)d_in[18];
  const float* conv2_bias = (const float*)d_in[19];
  const float* proj_w1    = (const float*)d_in[20];
  const float* proj_b1    = (const float*)d_in[21];
  const float* proj_w2    = (const float*)d_in[22];
  const float* proj_b2    = (const float*)d_in[23];
  const int*   edge_src   = (const int*)d_in[24];
  const int*   edge_dst   = (const int*)d_in[25];

  const int Nu = in_sizes[0] / 101;   // 200000 (multiple of 32)
  const int Ni = in_sizes[1];         // 20000  (multiple of 32)
  const int E  = in_sizes[24];        // 1000000

  // ---- workspace layout (256B aligned), with reuse: bigbuf = xl1 -> xr2 -> z1
  char*  ws  = (char*)d_ws;
  size_t off = 0;
  auto alloc = [&](size_t bytes) -> float* {
    float* p = (float*)(ws + off);
    off += (bytes + 255) & ~(size_t)255;
    return p;
  };
  float*    user_x = alloc((size_t)Nu * 32 * 4);    // 25.6 MB
  float*    item_x = alloc((size_t)Ni * 32 * 4);    //  2.6 MB
  float*    bigbuf = alloc((size_t)Nu * 128 * 4);   // 102.4 MB (xl1/xr2/z1)
  float*    xr1    = alloc((size_t)Ni * 128 * 4);   // 10.2 MB
  float*    elog   = alloc((size_t)E * 4 * 4);      // 16.0 MB
  unsigned* maxbuf = (unsigned*)alloc((size_t)Nu * 4 * 4); // 3.2 MB
  float*    den    = alloc((size_t)Nu * 4 * 4);     //  3.2 MB
  float*    acc1   = alloc((size_t)Ni * 128 * 4);   // 10.2 MB (-> item feat)
  float*    xl2    = alloc((size_t)Ni * 128 * 4);   // 10.2 MB

  float* user_out = (float*)d_out;                  // [Nu,128]
  float* z_out    = (float*)d_out + (size_t)Nu * 128;

  auto gemm = [&](const float* A, const float* B, const float* b, float* C,
                  int M, int N, int K, int relu) {
    if ((N & 63) == 0) {                       // N mult of 64 -> 32x64 blocks
      const int tiles = (M >> 5) * (N >> 6);
      wmma_gemm_f32<2, 4><<<(tiles + 7) / 8, 256, 0, stream>>>(A, B, b, C, M, N, K, relu);
    } else {                                   // N == 32 -> 32x32 blocks
      const int tiles = (M >> 5) * (N >> 5);
      wmma_gemm_f32<2, 2><<<(tiles + 7) / 8, 256, 0, stream>>>(A, B, b, C, M, N, K, relu);
    }
  };
  auto zero = [&](void* p, long n) {
    int blocks = (int)((n + 255) / 256);
    if (blocks > 4096) blocks = 4096;
    if (blocks < 1) blocks = 1;
    zero_f32<<<blocks, 256, 0, stream>>>((float*)p, n);
  };

  const int eb  = (E + 7) / 8;            // wave-per-edge kernels
  const int ehb = (E * 4 + 255) / 256;    // thread-per-(edge,head)

  // ---- input projections ------------------------------------------------
  gemm(customer_x, user_lin_w, user_lin_b, user_x, Nu, 32, 101, 0);
  gemm(fund_x,     item_lin_w, item_lin_b, item_x, Ni, 32, 1,   0);

  // ---- conv1: user -> item ---------------------------------------------
  gemm(user_x, conv1_wl, conv1_bl, bigbuf, Nu, 128, 32, 0);   // xl1
  gemm(item_x, conv1_wr, conv1_br, xr1,    Ni, 128, 32, 0);   // xr1
  zero(maxbuf, (long)Ni * 4);
  zero(den,    (long)Ni * 4);
  zero(acc1,   (long)Ni * 128);
  edge_logits_kernel<<<eb, 256, 0, stream>>>(bigbuf, xr1, edge_attr, conv1_we,
                                             conv1_att, edge_src, edge_dst,
                                             elog, maxbuf, E);
  edge_exp_kernel<<<ehb, 256, 0, stream>>>(elog, maxbuf, den, edge_dst, E);
  edge_agg_kernel<<<eb, 256, 0, stream>>>(bigbuf, elog, den, edge_src, edge_dst,
                                          acc1, E);
  bias_act_kernel<<<(int)(((long)Ni * 128 + 255) / 256), 256, 0, stream>>>(
      acc1, conv1_bias, (long)Ni * 128, 1);                   // item feat (ReLU)

  // ---- conv2: item -> user (flipped edges, no edge_attr) ----------------
  gemm(acc1,   conv2_wl, conv2_bl, xl2,    Ni, 128, 128, 0);  // xl2
  gemm(user_x, conv2_wr, conv2_br, bigbuf, Nu, 128, 32,  0);  // xr2
  zero(maxbuf,   (long)Nu * 4);
  zero(den,      (long)Nu * 4);
  zero(user_out, (long)Nu * 128);
  edge_logits_kernel<<<eb, 256, 0, stream>>>(xl2, bigbuf, nullptr, nullptr,
                                             conv2_att, edge_dst, edge_src,
                                             elog, maxbuf, E);
  edge_exp_kernel<<<ehb, 256, 0, stream>>>(elog, maxbuf, den, edge_src, E);
  edge_agg_kernel<<<eb, 256, 0, stream>>>(xl2, elog, den, edge_dst, edge_src,
                                          user_out, E);
  bias_act_kernel<<<(int)(((long)Nu * 128 + 255) / 256), 256, 0, stream>>>(
      user_out, conv2_bias, (long)Nu * 128, 0);               // user_out (no ReLU)

  // ---- projection head --------------------------------------------------
  gemm(user_out, proj_w1, proj_b1, bigbuf, Nu, 128, 128, 1);  // z1 = relu(...)
  gemm(bigbuf,   proj_w2, proj_b2, z_out,  Nu, 128, 128, 0);  // z
}